// AgentLoss_11261404250301
// MI455X (gfx1250) — compile-verified
//
#include <hip/hip_runtime.h>
#include <hip/hip_bf16.h>

#define EPS_   1e-5f
#define L_     4
#define B_     16
#define N_     1024
#define C_     64
#define ISTRIP 128
#define JSTRIP 64
#define LDPAD  68   // 64 floats + 4-float skew -> 4-bank rotation per row, conflict-free b64 reads

typedef __attribute__((ext_vector_type(2))) float v2f;
typedef __attribute__((ext_vector_type(8))) float v8f;
typedef __attribute__((ext_vector_type(4))) int   v4i;

#if defined(__gfx1250__) && \
    __has_builtin(__builtin_amdgcn_global_load_async_to_lds_b128) && \
    __has_builtin(__builtin_amdgcn_s_wait_asynccnt)
#define HAVE_ASYNC_LDS 1
typedef v4i __attribute__((address_space(1)))* g_v4i_p;  // global (AS1)
typedef v4i __attribute__((address_space(3)))* l_v4i_p;  // LDS    (AS3)
#else
#define HAVE_ASYNC_LDS 0
#endif

__device__ __forceinline__ void cp_b128(float* lds_dst, const float* gsrc) {
#if HAVE_ASYNC_LDS
    // Generic->AS casts via uintptr_t: low 32 bits of a generic LDS pointer are
    // the LDS offset; generic == global address for global pointers.
    __builtin_amdgcn_global_load_async_to_lds_b128(
        (g_v4i_p)(uintptr_t)gsrc,
        (l_v4i_p)(uintptr_t)lds_dst, 0, 0);
#else
    *(float4*)lds_dst = *(const float4*)gsrc;
#endif
}

__device__ __forceinline__ void cp_wait() {
#if HAVE_ASYNC_LDS
    __builtin_amdgcn_s_wait_asynccnt(0);
#endif
}

// Kernel 1: per-row L2 norms. One thread per (l,b,n) row of 64 floats.
__global__ void agent_norms_kernel(const float* __restrict__ A, float* __restrict__ norms) {
    int r = blockIdx.x * blockDim.x + threadIdx.x;
    const float* row = A + (size_t)r * C_;
    float s = 0.f;
#pragma unroll
    for (int i = 0; i < C_; i += 4) {
        float4 v = *(const float4*)(row + i);
        s += v.x * v.x + v.y * v.y + v.z * v.z + v.w * v.w;
    }
    norms[r] = sqrtf(s);
}

// Kernel 2: per (slice, 128-row i-strip): upper-triangular Gram tiles via fp32 WMMA.
// Symmetry: only tiles with Tj >= Ti are computed; Tj > Ti contributes x2.
__global__ void __launch_bounds__(512, 1)
agent_gram_kernel(const float* __restrict__ A, const float* __restrict__ norms,
                  float* __restrict__ out) {
    __shared__ float ldsI[ISTRIP * LDPAD];   // 34.0 KB
    __shared__ float ldsJ[JSTRIP * LDPAD];   // 17.0 KB
    __shared__ float ldsN[N_];               //  4.0 KB
    __shared__ float red[512];               //  2.0 KB

    const int slice = blockIdx.x >> 3;          // N_/ISTRIP == 8 strips per slice
    const int strip = blockIdx.x & 7;
    const float* As = A + (size_t)slice * N_ * C_;
    const float* Ns = norms + (size_t)slice * N_;
    const int t = threadIdx.x;

    // Stage per-slice norms (1024 floats / 512 threads).
    ldsN[t]       = Ns[t];
    ldsN[t + 512] = Ns[t + 512];

    // Stage i-strip: 128 rows x 64 floats = 2048 float4, 4 per thread (async -> LDS).
    const int iRow0 = strip * ISTRIP;
#pragma unroll
    for (int v = 0; v < 4; ++v) {
        int fi  = t + v * 512;        // float4 index
        int row = fi >> 4;            // 16 float4 per row
        int c4  = (fi & 15) << 2;
        cp_b128(&ldsI[row * LDPAD + c4], As + (size_t)(iRow0 + row) * C_ + c4);
    }
    cp_wait();
    __syncthreads();

    const int wave = t >> 5;          // 16 waves
    const int lane = t & 31;
    const int half = lane >> 4;       // K/M split of the wave
    const int l15  = lane & 15;

    float acc = 0.f;

    // j-strips below this block's i-strip are covered by other blocks (symmetry).
    for (int js = 2 * strip; js < N_ / JSTRIP; ++js) {
        // Stage j-strip: 64 rows x 64 floats = 1024 float4, 2 per thread.
#pragma unroll
        for (int v = 0; v < 2; ++v) {
            int fi  = t + v * 512;
            int row = fi >> 4;
            int c4  = (fi & 15) << 2;
            cp_b128(&ldsJ[row * LDPAD + c4], As + (size_t)(js * JSTRIP + row) * C_ + c4);
        }
        cp_wait();
        __syncthreads();

        // 8 i-tiles x 4 j-tiles = 32 tiles of 16x16; 16 waves -> 2 tiles each.
#pragma unroll
        for (int tt = 0; tt < 2; ++tt) {
            const int tile = wave + tt * 16;
            const int ti = tile >> 2;
            const int tj = tile & 3;
            const int Ti = strip * 8 + ti;     // global 16-row i-tile index
            const int Tj = js * 4 + tj;        // global 16-col j-tile index
            if (Tj < Ti) continue;             // wave-uniform skip: below diagonal
            const float wgt = (Tj > Ti) ? 2.0f : 1.0f;

            // A frag (16x4 f32): lane l15 = row M, half selects K pair {0,1} vs {2,3}.
            // B frag = A-rows(j): identical lane layout because G = A * A^T.
            const float* ai = &ldsI[(ti * 16 + l15) * LDPAD + 2 * half];
            const float* bj = &ldsJ[(tj * 16 + l15) * LDPAD + 2 * half];
            v8f c = {};
#pragma unroll
            for (int k = 0; k < 16; ++k) {     // K = 64 in 16 steps of 4
                v2f a, b;
                a.x = ai[4 * k];  a.y = ai[4 * k + 1];
                b.x = bj[4 * k];  b.y = bj[4 * k + 1];
                c = __builtin_amdgcn_wmma_f32_16x16x4_f32(
                        false, a, false, b, (short)0, c, false, false);
            }
            // C/D layout: element r -> M = r + 8*half, N = l15.
            const int jg = Tj * 16 + l15;      // slice-local column
            const float nj = ldsN[jg];
#pragma unroll
            for (int r = 0; r < 8; ++r) {
                const int ig = Ti * 16 + r + 8 * half;
                const float ni = ldsN[ig];
                const float inv = __builtin_amdgcn_rcpf(fmaf(ni, nj, EPS_)); // v_rcp_f32
                const float s = c[r] * inv * wgt;
                acc += (ig != jg) ? s : 0.f;   // zero diagonal, branchless
            }
        }
        __syncthreads();   // before reusing ldsJ
    }

    // Block reduction + scaled atomic accumulate.
    red[t] = acc;
    __syncthreads();
    for (int s = 256; s > 0; s >>= 1) {
        if (t < s) red[t] += red[t + s];
        __syncthreads();
    }
    if (t == 0) {
        const float scale = 1.0f / ((float)N_ * (float)(N_ - 1) * (float)B_);
        atomicAdd(out, red[0] * scale);
    }
}

extern "C" void kernel_launch(void* const* d_in, const int* in_sizes, int n_in,
                              void* d_out, int out_size, void* d_ws, size_t ws_size,
                              hipStream_t stream) {
    const float* A  = (const float*)d_in[0];
    float* out      = (float*)d_out;
    float* norms    = (float*)d_ws;   // 65536 floats = 256 KB scratch

    (void)hipMemsetAsync(d_out, 0, sizeof(float) * (size_t)out_size, stream);
    agent_norms_kernel<<<(L_ * B_ * N_) / 256, 256, 0, stream>>>(A, norms);
    agent_gram_kernel<<<L_ * B_ * (N_ / ISTRIP), 512, 0, stream>>>(A, norms, out);
}